// DeeperGCN_62096637165587
// MI455X (gfx1250) — compile-verified
//
#include <hip/hip_runtime.h>
#include <hip/hip_bf16.h>

#define Nn 10000
#define Ee 640000
#define Hh 128
#define XDim 8
#define EDim 8
#define YDim 112
#define Ll 4
#define NH (Nn * Hh)

typedef float v2f __attribute__((ext_vector_type(2)));
typedef float v8f __attribute__((ext_vector_type(8)));

__device__ __forceinline__ v8f wmma4(v2f a, v2f b, v8f c) {
    // D(16x16,f32) = A(16x4,f32) * B(4x16,f32) + C
    return __builtin_amdgcn_wmma_f32_16x16x4_f32(
        /*neg_a=*/false, a, /*neg_b=*/false, b,
        /*c_mod=*/(short)0, c, /*reuse_a=*/false, /*reuse_b=*/false);
}

// -------- node encoder: h = x @ Wn + bn  ([N,8] @ [8,128]) --------
__global__ void encode_nodes(const float* __restrict__ x, const float* __restrict__ Wn,
                             const float* __restrict__ bn, float* __restrict__ h) {
    int idx = blockIdx.x * blockDim.x + threadIdx.x;
    if (idx >= NH) return;
    int n = idx >> 7, c = idx & 127;
    float acc = bn[c];
#pragma unroll
    for (int k = 0; k < XDim; ++k) acc += x[n * XDim + k] * Wn[k * Hh + c];
    h[idx] = acc;
}

// -------- per-layer init: mx=0 (logits>0), den=0, agg=hin (root add), stats=0 --------
__global__ void init_layer(const float* __restrict__ hin, float* __restrict__ mx,
                           float* __restrict__ den, float* __restrict__ agg,
                           float* __restrict__ stats) {
    int idx = blockIdx.x * blockDim.x + threadIdx.x;
    if (idx < NH) {
        mx[idx] = 0.0f;
        den[idx] = 0.0f;
        agg[idx] = hin[idx];
    }
    if (blockIdx.x == 0 && threadIdx.x < 2) stats[threadIdx.x] = 0.0f;
}

// -------- edge passes: recompute ea = edge_attr@We + be on the fly (L2-resident) --------
// PASS 0: segment-max(logit) via uint atomicMax (logits > 0)
// PASS 1: den += exp(logit - mx)
// PASS 2: agg += (ex / (den+eps)) * msg
template <int PASS>
__global__ void edge_pass(const int* __restrict__ ei, const float* __restrict__ eattr,
                          const float* __restrict__ We, const float* __restrict__ be,
                          const float* __restrict__ hin, const float* __restrict__ t, int layer,
                          float* __restrict__ mx, float* __restrict__ den,
                          float* __restrict__ agg) {
    int e = blockIdx.x * 2 + (threadIdx.x >> 7);  // 256 threads = 2 edges x 128 channels
    int ch = threadIdx.x & 127;
    if (e >= Ee) return;
    int src = ei[e];
    int dst = ei[Ee + e];
    float ea = be[ch];
#pragma unroll
    for (int k = 0; k < EDim; ++k) ea += eattr[e * EDim + k] * We[k * Hh + ch];
    float msg = fmaxf(hin[src * Hh + ch] + ea, 0.0f) + 1e-7f;
    float logit = msg * t[layer];
    int o = dst * Hh + ch;
    if (PASS == 0) {
        atomicMax((unsigned int*)mx + o, __float_as_uint(logit));
    } else {
        float ex = expf(logit - mx[o]);
        if (PASS == 1) {
            atomicAdd(den + o, ex);
        } else {
            float alpha = ex / (den[o] + 1e-16f);
            atomicAdd(agg + o, alpha * msg);
        }
    }
}

// -------- per-node LayerNorm + ReLU (128 ch per node, one block per node) --------
__global__ void node_ln_relu(const float* __restrict__ x, const float* __restrict__ g,
                             const float* __restrict__ b, float* __restrict__ out) {
    __shared__ float s1[128], s2[128];
    int n = blockIdx.x, c = threadIdx.x;
    float v = x[n * Hh + c];
    s1[c] = v;
    s2[c] = v * v;
    __syncthreads();
    for (int off = 64; off > 0; off >>= 1) {
        if (c < off) {
            s1[c] += s1[c + off];
            s2[c] += s2[c + off];
        }
        __syncthreads();
    }
    float mean = s1[0] * (1.0f / Hh);
    float var = s2[0] * (1.0f / Hh) - mean * mean;
    float rstd = rsqrtf(var + 1e-5f);
    out[n * Hh + c] = fmaxf((v - mean) * rstd * g[c] + b[c], 0.0f);
}

// -------- WMMA GEMM: C[M,NC] (+)= op_A(A[M,K]) @ W[K,NC] + bias --------
// One wave per 16x16 tile. DO_STATS: accumulate sum/sumsq of outputs (for graph_ln).
// LN_A: apply graph_ln (per-column scale/shift from stats) + ReLU to A loads.
template <int DO_STATS, int LN_A>
__global__ void __launch_bounds__(32) wmma_gemm(
    const float* __restrict__ A, const float* __restrict__ W, const float* __restrict__ bias,
    float* __restrict__ C, int K_, int NC, const float* __restrict__ lg,
    const float* __restrict__ lb, const float* __restrict__ stats, float inv_count,
    float* __restrict__ stats_out, int residual) {
    int row0 = blockIdx.x * 16;
    int col0 = blockIdx.y * 16;
    int lane = threadIdx.x;
    int hl = lane >> 4;   // half of wave: K sub-pair select
    int l15 = lane & 15;

    float mean = 0.0f, rstd = 0.0f;
    if (LN_A) {
        float s = stats[0] * inv_count;
        float sq = stats[1] * inv_count;
        mean = s;
        float var = sq - s * s;
        rstd = 1.0f / (sqrtf(var) + 1e-5f);  // pyg graph LN: xc / (std + eps)
    }

    v8f c = {};
    int arow = row0 + l15;   // A layout: lanes 0-15 and 16-31 both carry M=0..15
    int bcol = col0 + l15;   // B layout: row striped across lanes within a VGPR
    for (int k = 0; k < K_; k += 4) {
        int ka = k + hl * 2;
        float a0 = A[arow * K_ + ka];
        float a1 = A[arow * K_ + ka + 1];
        if (LN_A) {
            a0 = fmaxf((a0 - mean) * rstd * lg[ka] + lb[ka], 0.0f);
            a1 = fmaxf((a1 - mean) * rstd * lg[ka + 1] + lb[ka + 1], 0.0f);
        }
        v2f a = {a0, a1};
        v2f b = {W[ka * NC + bcol], W[(ka + 1) * NC + bcol]};
        c = wmma4(a, b, c);
    }

    float bc = bias[bcol];
    float s = 0.0f, sq = 0.0f;
#pragma unroll
    for (int r = 0; r < 8; ++r) {
        int m = row0 + r + hl * 8;  // C/D layout: lanes 16-31 hold M = r+8
        float v = c[r] + bc;
        int o = m * NC + bcol;
        if (residual) C[o] += v; else C[o] = v;
        if (DO_STATS) { s += v; sq += v * v; }
    }
    if (DO_STATS) {
        for (int off = 16; off > 0; off >>= 1) {
            s += __shfl_xor(s, off, 32);
            sq += __shfl_xor(sq, off, 32);
        }
        if (lane == 0) {
            atomicAdd(stats_out, s);
            atomicAdd(stats_out + 1, sq);
        }
    }
}

extern "C" void kernel_launch(void* const* d_in, const int* in_sizes, int n_in,
                              void* d_out, int out_size, void* d_ws, size_t ws_size,
                              hipStream_t stream) {
    const float* x    = (const float*)d_in[0];
    const int*   ei   = (const int*)d_in[1];
    const float* eatt = (const float*)d_in[2];
    const float* Wn   = (const float*)d_in[3];
    const float* bn   = (const float*)d_in[4];
    const float* We   = (const float*)d_in[5];
    const float* be   = (const float*)d_in[6];
    const float* t    = (const float*)d_in[7];
    const float* W1   = (const float*)d_in[8];
    const float* b1   = (const float*)d_in[9];
    const float* g1   = (const float*)d_in[10];
    const float* bt1  = (const float*)d_in[11];
    const float* W2   = (const float*)d_in[12];
    const float* b2   = (const float*)d_in[13];
    const float* lng  = (const float*)d_in[14];
    const float* lnb  = (const float*)d_in[15];
    const float* Wout = (const float*)d_in[16];
    const float* bout = (const float*)d_in[17];
    float* out = (float*)d_out;

    float* ws  = (float*)d_ws;
    float* h   = ws;            // [N,H]
    float* r   = h + NH;        // [N,H]
    float* mx  = r + NH;        // [N,H]
    float* den = mx + NH;       // [N,H]
    float* agg = den + NH;      // [N,H]
    float* u   = agg + NH;      // [N,2H]
    float* st  = u + (size_t)Nn * 2 * Hh;  // 2 scalars: sum, sumsq

    const int tpb = 256;
    const dim3 g_edges(Ee / 2);
    const float inv_cnt = 1.0f / (float)(Nn * 2 * Hh);

    encode_nodes<<<(NH + tpb - 1) / tpb, tpb, 0, stream>>>(x, Wn, bn, h);

    for (int i = 0; i < Ll; ++i) {
        const float* hin = h;
        if (i > 0) {
            node_ln_relu<<<Nn, Hh, 0, stream>>>(h, lng + i * Hh, lnb + i * Hh, r);
            hin = r;
        }
        init_layer<<<(NH + tpb - 1) / tpb, tpb, 0, stream>>>(hin, mx, den, agg, st);
        edge_pass<0><<<g_edges, tpb, 0, stream>>>(ei, eatt, We, be, hin, t, i, mx, den, agg);
        edge_pass<1><<<g_edges, tpb, 0, stream>>>(ei, eatt, We, be, hin, t, i, mx, den, agg);
        edge_pass<2><<<g_edges, tpb, 0, stream>>>(ei, eatt, We, be, hin, t, i, mx, den, agg);

        // u = agg @ W1[i] + b1[i]  (+ global stats for graph_ln)
        wmma_gemm<1, 0><<<dim3(Nn / 16, (2 * Hh) / 16), 32, 0, stream>>>(
            agg, W1 + (size_t)i * Hh * 2 * Hh, b1 + i * 2 * Hh, u, Hh, 2 * Hh,
            nullptr, nullptr, nullptr, 0.0f, st, 0);

        // h (+)= relu(graph_ln(u)) @ W2[i] + b2[i]
        wmma_gemm<0, 1><<<dim3(Nn / 16, Hh / 16), 32, 0, stream>>>(
            u, W2 + (size_t)i * 2 * Hh * Hh, b2 + i * Hh, h, 2 * Hh, Hh,
            g1 + i * 2 * Hh, bt1 + i * 2 * Hh, st, inv_cnt, nullptr, (i > 0) ? 1 : 0);
    }

    // final: r = relu(node_ln(h, lng[0], lnb[0])); out = r @ Wout + bout
    node_ln_relu<<<Nn, Hh, 0, stream>>>(h, lng, lnb, r);
    wmma_gemm<0, 0><<<dim3(Nn / 16, YDim / 16), 32, 0, stream>>>(
        r, Wout, bout, out, Hh, YDim, nullptr, nullptr, nullptr, 0.0f, nullptr, 0);
}